// ASNN_78451872629444
// MI455X (gfx1250) — compile-verified
//
#include <hip/hip_runtime.h>

typedef __attribute__((ext_vector_type(2))) float v2f;
typedef __attribute__((ext_vector_type(8))) float v8f;

#define B_    512
#define T_    512
#define IND_  64
#define WID_  128
#define BR_   16     // batch rows per workgroup
#define LDSTR 132    // padded LDS row stride (132 % 64 = 4 -> conflict-free column reads)
#define GAMMA 0.01f
#define EPS   0.01f

// Workgroup split-barrier that only waits for LDS visibility (dscnt), NOT for
// global stores/loads like __syncthreads() does. Output NT stores drain in the
// background; the x-prefetch loads stay in flight across the barrier.
__device__ __forceinline__ void lds_barrier() {
    asm volatile("s_wait_dscnt 0x0\n\t"
                 "s_barrier_signal -1\n\t"
                 "s_barrier_wait -1" ::: "memory");
}

__device__ __forceinline__ float fast_tanh(float v) {
#if __has_builtin(__builtin_amdgcn_tanhf)
    return __builtin_amdgcn_tanhf(v);   // v_tanh_f32
#else
    return tanhf(v);
#endif
}

// Load the 16 A-operand chunks of x[:, t, :] for this lane (NT: read-once stream).
__device__ __forceinline__ void load_x(v2f (&xr)[16], const float* __restrict__ xrow,
                                       int t, int hi) {
    #pragma unroll
    for (int c = 0; c < 16; ++c) {
        xr[c] = __builtin_nontemporal_load(
                    (const v2f*)(xrow + (size_t)t * IND_ + 4 * c + 2 * hi));
    }
}

__global__ __launch_bounds__(256, 1)
void asnn_scan_kernel(const float* __restrict__ x,
                      const float* __restrict__ init_state,
                      const float* __restrict__ W,
                      const float* __restrict__ V,
                      const float* __restrict__ b,
                      float* __restrict__ out)
{
    // ping-pong state buffers: BR_ rows x WID_ cols, padded stride
    __shared__ float st[2][BR_ * LDSTR];

    const int tid   = threadIdx.x;
    const int wave  = tid >> 5;          // 8 waves, each owns 16 output columns
    const int lane  = tid & 31;
    const int hi    = lane >> 4;         // upper half-wave flag
    const int ln    = lane & 15;
    const int col   = wave * 16 + ln;    // this lane's N column (C/D layout)
    const int bbase = blockIdx.x * BR_;

    // ---- Static B-operands in registers, WMMA B layout for 16x16x4:
    // VGPR0: lanes0-15 = row K=4c+0, lanes16-31 = row K=4c+2 ; VGPR1: +1 row.
    // A = W - W^T built directly from global W (resident in L2).
    v2f Ab[32];
    #pragma unroll
    for (int c = 0; c < 32; ++c) {
        const int k = 4 * c + 2 * hi;
        Ab[c].x = W[(k + 0) * WID_ + col] - W[col * WID_ + (k + 0)];
        Ab[c].y = W[(k + 1) * WID_ + col] - W[col * WID_ + (k + 1)];
    }
    v2f Vb[16];
    #pragma unroll
    for (int c = 0; c < 16; ++c) {
        const int k = 4 * c + 2 * hi;
        Vb[c].x = V[(k + 0) * WID_ + col];
        Vb[c].y = V[(k + 1) * WID_ + col];
    }
    const float bias = b[col];

    // ---- Initial state: own 16x16 tile kept in C/D register layout + published to LDS.
    v8f s;
    #pragma unroll
    for (int r = 0; r < 8; ++r) {
        const int m = r + 8 * hi;
        const float v = init_state[(size_t)(bbase + m) * WID_ + col];
        s[r] = v;
        st[0][m * LDSTR + col] = v;
    }
    lds_barrier();

    // Row of x feeding this lane's A-operand M index (M = ln for both half-waves).
    const float* xrow = x + (size_t)(bbase + ln) * T_ * IND_;
    // Per-lane base of this lane's state row in each LDS buffer.
    const float* srow0 = &st[0][ln * LDSTR];
    const float* srow1 = &st[1][ln * LDSTR];

    // Register double-buffer for x operands; prefetch t=0.
    v2f xA[16], xB[16];
    load_x(xA, xrow, 0, hi);

    #define ASNN_STEP(T_IDX, XCUR, XNEXT)                                          \
    {                                                                              \
        const int t  = (T_IDX);                                                    \
        const int tn = (t + 1 < T_) ? (t + 1) : (T_ - 1);                          \
        const float* stc = (t & 1) ? srow1 : srow0;                                \
                                                                                   \
        /* prefetch first half of state A-operands; latency covered by V WMMAs */  \
        v2f sa[16];                                                                \
        _Pragma("unroll")                                                          \
        for (int c = 0; c < 16; ++c) {                                             \
            const int k = 4 * c + 2 * hi;                                          \
            sa[c].x = stc[k + 0];                                                  \
            sa[c].y = stc[k + 1];                                                  \
        }                                                                          \
        /* prefetch x for next step; latency covered by this whole step */         \
        load_x(XNEXT, xrow, tn, hi);                                               \
                                                                                   \
        v8f acc0 = {bias, bias, bias, bias, bias, bias, bias, bias};               \
        v8f acc1 = {};                                                             \
                                                                                   \
        /* x_t @ V first: operands already resident (prefetched last step) */      \
        _Pragma("unroll")                                                          \
        for (int c = 0; c < 16; ++c) {                                             \
            if (c & 1)                                                             \
                acc1 = __builtin_amdgcn_wmma_f32_16x16x4_f32(                      \
                           false, XCUR[c], false, Vb[c], (short)0, acc1,           \
                           false, false);                                          \
            else                                                                   \
                acc0 = __builtin_amdgcn_wmma_f32_16x16x4_f32(                      \
                           false, XCUR[c], false, Vb[c], (short)0, acc0,           \
                           false, false);                                          \
        }                                                                          \
                                                                                   \
        /* state @ A, first half from the prefetched registers */                  \
        _Pragma("unroll")                                                          \
        for (int c = 0; c < 16; ++c) {                                             \
            if (c & 1)                                                             \
                acc1 = __builtin_amdgcn_wmma_f32_16x16x4_f32(                      \
                           false, sa[c], false, Ab[c], (short)0, acc1,             \
                           false, false);                                          \
            else                                                                   \
                acc0 = __builtin_amdgcn_wmma_f32_16x16x4_f32(                      \
                           false, sa[c], false, Ab[c], (short)0, acc0,             \
                           false, false);                                          \
        }                                                                          \
        /* second half: ds loads pipeline behind the WMMAs above */                \
        _Pragma("unroll")                                                          \
        for (int c = 16; c < 32; ++c) {                                            \
            const int k = 4 * c + 2 * hi;                                          \
            v2f a;                                                                 \
            a.x = stc[k + 0];                                                      \
            a.y = stc[k + 1];                                                      \
            if (c & 1)                                                             \
                acc1 = __builtin_amdgcn_wmma_f32_16x16x4_f32(                      \
                           false, a, false, Ab[c], (short)0, acc1, false, false);  \
            else                                                                   \
                acc0 = __builtin_amdgcn_wmma_f32_16x16x4_f32(                      \
                           false, a, false, Ab[c], (short)0, acc0, false, false);  \
        }                                                                          \
                                                                                   \
        /* elementwise update in C/D layout; publish new state to LDS */           \
        float* stn = st[(t + 1) & 1];                                              \
        _Pragma("unroll")                                                          \
        for (int r = 0; r < 8; ++r) {                                              \
            const int m     = r + 8 * hi;                                          \
            const float pre = acc0[r] + acc1[r] - GAMMA * s[r];                    \
            s[r] = s[r] + EPS * fast_tanh(pre);                                    \
            stn[m * LDSTR + col] = s[r];                                           \
        }                                                                          \
        lds_barrier();  /* waits dscnt only: NT stores & prefetch stay in flight */\
        /* stream outputs after the barrier, off the inter-wave critical path */   \
        _Pragma("unroll")                                                          \
        for (int r = 0; r < 8; ++r) {                                              \
            const int m = r + 8 * hi;                                              \
            __builtin_nontemporal_store(                                           \
                s[r], out + ((size_t)(bbase + m) * T_ + t) * WID_ + col);          \
        }                                                                          \
    }

    for (int t2 = 0; t2 < T_; t2 += 2) {
        ASNN_STEP(t2 + 0, xA, xB)   // reads xA, prefetches t2+1 into xB
        ASNN_STEP(t2 + 1, xB, xA)   // reads xB, prefetches t2+2 into xA
    }
    #undef ASNN_STEP

    // final_state appended after outputs: [B, WID_]
    float* fin = out + (size_t)B_ * T_ * WID_;
    #pragma unroll
    for (int r = 0; r < 8; ++r) {
        const int m = r + 8 * hi;
        fin[(size_t)(bbase + m) * WID_ + col] = s[r];
    }
}

extern "C" void kernel_launch(void* const* d_in, const int* in_sizes, int n_in,
                              void* d_out, int out_size, void* d_ws, size_t ws_size,
                              hipStream_t stream) {
    (void)in_sizes; (void)n_in; (void)d_ws; (void)ws_size; (void)out_size;
    const float* x   = (const float*)d_in[0];
    const float* ist = (const float*)d_in[1];
    const float* W   = (const float*)d_in[2];
    const float* V   = (const float*)d_in[3];
    const float* b   = (const float*)d_in[4];
    float* out       = (float*)d_out;

    dim3 grid(B_ / BR_);   // 32 workgroups
    dim3 block(256);       // 8 wave32s
    asnn_scan_kernel<<<grid, block, 0, stream>>>(x, ist, W, V, b, out);
}